// MMDLoss_74603581931591
// MI455X (gfx1250) — compile-verified
//
#include <hip/hip_runtime.h>

typedef __attribute__((ext_vector_type(2))) float v2f;
typedef __attribute__((ext_vector_type(8))) float v8f;
typedef __attribute__((ext_vector_type(4))) unsigned int v4u;
typedef __attribute__((ext_vector_type(8))) int v8i;
typedef __attribute__((ext_vector_type(4))) int v4i;

#define N_PTS 8192
#define DIMS 256
#define BLK 128
#define KC 32
#define LDS_STRIDE 36   // 32 data floats + 4 pad floats per row (TDM pad: interval=32DW, amount=4DW)
#define NCHUNK (DIMS / KC)

// ---------------- row squared-norms: one wave per row ----------------
__global__ __launch_bounds__(256) void norms_kernel(const float* __restrict__ X,
                                                    float* __restrict__ norms) {
    const int lane = threadIdx.x & 31;
    const int wave = threadIdx.x >> 5;
    const int row  = blockIdx.x * 8 + wave;
    const float* p = X + (size_t)row * DIMS;
    float s = 0.f;
#pragma unroll
    for (int i = 0; i < DIMS; i += 32) {
        float v = p[i + lane];
        s += v * v;
    }
#pragma unroll
    for (int off = 16; off > 0; off >>= 1) s += __shfl_down(s, off, 32);
    if (lane == 0) norms[row] = s;
}

// ---------------- TDM: DMA one 128x32 f32 tile (row stride 256) into LDS ----------------
// LDS rows padded 32->36 floats via D# pad (interval code 4 = 32 DWORDs, amount code 3 = 4 DWORDs).
__device__ __forceinline__ void tdm_load_tile_f32(const float* gptr, unsigned lds_off) {
    unsigned long long ga = (unsigned long long)(uintptr_t)gptr;
    v4u g0;
    g0.x = 1u;                                   // count=1, user descriptor
    g0.y = lds_off;                              // lds_addr (bytes)
    g0.z = (unsigned)ga;                         // global_addr[31:0]
    g0.w = (unsigned)((ga >> 32) & 0x01FFFFFFull) | (2u << 30);  // global_addr[56:32] | type=2
    v8i g1;
    g1[0] = (int)((2u << 16) | (1u << 20) | (4u << 22) | (3u << 25)); // data_size=4B, pad_en, interval=32DW, amount=4DW
    g1[1] = (int)((unsigned)DIMS << 16);         // tensor_dim0[15:0]=256
    g1[2] = (int)((unsigned)N_PTS << 16);        // tensor_dim0 hi=0 | tensor_dim1[15:0]=8192
    g1[3] = (int)((unsigned)KC << 16);           // tensor_dim1 hi=0 | tile_dim0=32
    g1[4] = BLK;                                 // tile_dim1=128, tile_dim2=0
    g1[5] = DIMS;                                // tensor_dim0_stride[31:0]=256
    g1[6] = 0;                                   // stride0 hi | tensor_dim1_stride lo
    g1[7] = 0;
    v4i gz4 = {0, 0, 0, 0};                      // 2D tensor: groups 2/3 unused
    v8i gz8 = {0, 0, 0, 0, 0, 0, 0, 0};
    __builtin_amdgcn_tensor_load_to_lds(g0, g1, gz4, gz4, gz8, 0);
}

// ---------------- 128x128 Gram tile -> exp -> partial sum ----------------
// 256 threads = 8 waves, 2(M) x 4(N); each wave: 64x32 subtile = 4x2 WMMA accs.
// Tiles staged by the Tensor Data Mover, double buffered.
__global__ __launch_bounds__(256) void gram_kernel(const float* __restrict__ X,
                                                   const float* __restrict__ Y,
                                                   const float* __restrict__ nx,
                                                   const float* __restrict__ ny,
                                                   float* __restrict__ partials,
                                                   float weight) {
    __shared__ float xs[2][BLK * LDS_STRIDE];
    __shared__ float ys[2][BLK * LDS_STRIDE];
    __shared__ float nxs[BLK];
    __shared__ float nys[BLK];
    __shared__ float red[8];

    const int tid   = threadIdx.x;
    const int lane  = tid & 31;
    const int wave  = tid >> 5;       // 0..7
    const int waveM = wave >> 2;      // 0..1 -> 64-row half
    const int waveN = wave & 3;       // 0..3 -> 32-col quarter
    const int rowBase = blockIdx.y * BLK;
    const int colBase = blockIdx.x * BLK;

    if (tid < BLK) nxs[tid]       = nx[rowBase + tid];
    else           nys[tid - BLK] = ny[colBase + tid - BLK];

    // prologue: wave 0 kicks off chunk 0 DMAs
    if (wave == 0) {
        tdm_load_tile_f32(X + (size_t)rowBase * DIMS, (unsigned)(uintptr_t)&xs[0][0]);
        tdm_load_tile_f32(Y + (size_t)colBase * DIMS, (unsigned)(uintptr_t)&ys[0][0]);
    }

    v8f zero = {};
    v8f acc[4][2];
#pragma unroll
    for (int m = 0; m < 4; ++m)
#pragma unroll
        for (int n = 0; n < 2; ++n) acc[m][n] = zero;

    const int lrow  = lane & 15;        // fragment row within 16x16 tile
    const int lkoff = (lane >> 4) * 2;  // K sub-offset per half-wave (ISA 16x4 f32 layout)

    for (int c = 0; c < NCHUNK; ++c) {
        const int cur = c & 1;
        if (wave == 0) {
            if (c + 1 < NCHUNK) {
                // prefetch next chunk into the other buffer (freed by previous trailing barrier)
                const int k0n = (c + 1) * KC;
                const int nxt = (c + 1) & 1;
                tdm_load_tile_f32(X + (size_t)rowBase * DIMS + k0n, (unsigned)(uintptr_t)&xs[nxt][0]);
                tdm_load_tile_f32(Y + (size_t)colBase * DIMS + k0n, (unsigned)(uintptr_t)&ys[nxt][0]);
                __builtin_amdgcn_s_wait_tensorcnt(2);  // in-order: <=2 outstanding => chunk c landed
            } else {
                __builtin_amdgcn_s_wait_tensorcnt(0);
            }
        }
        __syncthreads();   // chunk c visible to all waves

        // 8 K-substeps of 4 -> 64 fp32 WMMAs per wave per chunk
#pragma unroll
        for (int kk = 0; kk < 8; ++kk) {
            const int kcol = kk * 4 + lkoff;
            v2f a[4], b[2];
#pragma unroll
            for (int m = 0; m < 4; ++m)
                a[m] = *(const v2f*)(&xs[cur][(waveM * 64 + m * 16 + lrow) * LDS_STRIDE + kcol]);
#pragma unroll
            for (int n = 0; n < 2; ++n)
                b[n] = *(const v2f*)(&ys[cur][(waveN * 32 + n * 16 + lrow) * LDS_STRIDE + kcol]);
#pragma unroll
            for (int m = 0; m < 4; ++m)
#pragma unroll
                for (int n = 0; n < 2; ++n)
                    acc[m][n] = __builtin_amdgcn_wmma_f32_16x16x4_f32(
                        false, a[m], false, b[n], (short)0, acc[m][n], false, false);
        }
        __syncthreads();   // all waves done with buffer `cur` before it is DMA-overwritten
    }

    // epilogue: sq = nx + ny - 2*dot ; k = exp(-sq / D^2) ; accumulate
    const float invD2 = 1.0f / ((float)DIMS * (float)DIMS);
    const int mrowOff = (lane >> 4) * 8;   // C layout: lanes 16-31 hold M = v+8
    float lsum = 0.f;
#pragma unroll
    for (int m = 0; m < 4; ++m) {
#pragma unroll
        for (int n = 0; n < 2; ++n) {
            float nb = nys[waveN * 32 + n * 16 + (lane & 15)];
#pragma unroll
            for (int v = 0; v < 8; ++v) {
                float na = nxs[waveM * 64 + m * 16 + mrowOff + v];
                float sq = na + nb - 2.0f * acc[m][n][v];
                lsum += __expf(-sq * invD2);
            }
        }
    }
#pragma unroll
    for (int off = 16; off > 0; off >>= 1) lsum += __shfl_down(lsum, off, 32);
    if (lane == 0) red[wave] = lsum;
    __syncthreads();
    if (tid == 0) {
        float s = 0.f;
#pragma unroll
        for (int w = 0; w < 8; ++w) s += red[w];
        partials[blockIdx.y * gridDim.x + blockIdx.x] = s * weight;
    }
}

// ---------------- deterministic final reduction ----------------
__global__ __launch_bounds__(256) void reduce_kernel(const float* __restrict__ partials,
                                                     float* __restrict__ out, int count) {
    __shared__ float red[256];
    float s = 0.f;
    for (int i = threadIdx.x; i < count; i += 256) s += partials[i];
    red[threadIdx.x] = s;
    __syncthreads();
    for (int stride = 128; stride > 0; stride >>= 1) {
        if (threadIdx.x < stride) red[threadIdx.x] += red[threadIdx.x + stride];
        __syncthreads();
    }
    if (threadIdx.x == 0) out[0] = red[0];
}

extern "C" void kernel_launch(void* const* d_in, const int* in_sizes, int n_in,
                              void* d_out, int out_size, void* d_ws, size_t ws_size,
                              hipStream_t stream) {
    const float* A = (const float*)d_in[0];
    const float* B = (const float*)d_in[1];
    float* ws       = (float*)d_ws;
    float* nA       = ws;                 // 8192
    float* nB       = ws + N_PTS;         // 8192
    float* partials = ws + 2 * N_PTS;     // 3 * 4096
    float* out      = (float*)d_out;

    norms_kernel<<<N_PTS / 8, 256, 0, stream>>>(A, nA);
    norms_kernel<<<N_PTS / 8, 256, 0, stream>>>(B, nB);

    dim3 grid(N_PTS / BLK, N_PTS / BLK);   // 64 x 64
    const float invNN = 1.0f / ((float)N_PTS * (float)N_PTS);
    gram_kernel<<<grid, 256, 0, stream>>>(A, A, nA, nA, partials,        invNN);
    gram_kernel<<<grid, 256, 0, stream>>>(B, B, nB, nB, partials + 4096, invNN);
    gram_kernel<<<grid, 256, 0, stream>>>(A, B, nA, nB, partials + 8192, -2.0f * invNN);

    reduce_kernel<<<1, 256, 0, stream>>>(partials, out, 3 * 4096);
}